// centercompute_val_38027640439209
// MI455X (gfx1250) — compile-verified
//
#include <hip/hip_runtime.h>
#include <math.h>

#define NUM_CLASSES 3
#define DIM 128
#define EPSF 1e-12f

typedef float v2f __attribute__((ext_vector_type(2)));
typedef float v8f __attribute__((ext_vector_type(8)));

// d_ws layout (floats): [0 .. 3*128-1] class sums, [384..386] class counts
#define WS_FLOATS (NUM_CLASSES * DIM + NUM_CLASSES)

__global__ void zero_ws_kernel(float* __restrict__ ws) {
    int i = blockIdx.x * blockDim.x + threadIdx.x;
    if (i < WS_FLOATS) ws[i] = 0.0f;
}

// One-hot x features segment-sum via V_WMMA_F32_16X16X4_F32.
// Each wave processes groups of 4 samples; 8 WMMAs cover all 128 dims.
__global__ __launch_bounds__(256) void segsum_wmma_kernel(
    const float* __restrict__ features,
    const int*   __restrict__ labels,
    float*       __restrict__ ws,
    int N)
{
    __shared__ float sSums[NUM_CLASSES * DIM];
    __shared__ float sCnt[NUM_CLASSES];

    for (int i = threadIdx.x; i < NUM_CLASSES * DIM; i += blockDim.x) sSums[i] = 0.0f;
    if (threadIdx.x < NUM_CLASSES) sCnt[threadIdx.x] = 0.0f;
    __syncthreads();

    const int lane = threadIdx.x & 31;
    const int wid  = threadIdx.x >> 5;
    const int col  = lane & 15;   // N index (feature dim within 16-wide tile) / class M for A
    const int half = lane >> 4;   // selects K pair {0,1} vs {2,3}

    const int wavesPerBlock = blockDim.x >> 5;
    const int waveId   = blockIdx.x * wavesPerBlock + wid;
    const int numWaves = gridDim.x * wavesPerBlock;
    const int G = (N + 3) >> 2;   // groups of 4 samples

    v8f acc[8];
#pragma unroll
    for (int t = 0; t < 8; ++t) acc[t] = (v8f){0.f,0.f,0.f,0.f,0.f,0.f,0.f,0.f};
    float cntacc = 0.0f;

#if defined(__gfx1250__)
    for (int g = waveId; g < G; g += numWaves) {   // wave-uniform loop: EXEC stays all-1s
        const int s  = g << 2;
        const int r0 = s + 2 * half;        // sample index for K = 0 / 2
        const int r1 = s + 1 + 2 * half;    // sample index for K = 1 / 3
        const bool v0 = (r0 < N);
        const bool v1 = (r1 < N);
        const int cr0 = v0 ? r0 : 0;        // clamp (A=0 kills any garbage in B)
        const int cr1 = v1 ? r1 : 0;

        const int l0 = labels[cr0];
        const int l1 = labels[cr1];

        // A-matrix (16x4 one-hot, f32): VGPR0 = K(0|2), VGPR1 = K(1|3); M = col
        v2f A;
        A.x = (v0 && (l0 == col)) ? 1.0f : 0.0f;
        A.y = (v1 && (l1 == col)) ? 1.0f : 0.0f;
        cntacc += A.x + A.y;

        const float* p0 = features + (long long)cr0 * DIM + col;
        const float* p1 = features + (long long)cr1 * DIM + col;

#pragma unroll
        for (int t = 0; t < 8; ++t) {
            // B-matrix (4x16, f32): row K striped across lanes; dims 16t..16t+15
            v2f B;
            B.x = p0[t * 16];
            B.y = p1[t * 16];
            acc[t] = __builtin_amdgcn_wmma_f32_16x16x4_f32(
                /*neg_a=*/false, A, /*neg_b=*/false, B,
                /*c_mod=*/(short)0, acc[t],
                /*reuse_a=*/false, /*reuse_b=*/false);
        }
    }
#endif

    // C/D layout: lanes 0-15, VGPR c -> class M=c, dim N = 16t + col
    if (half == 0) {
#pragma unroll
        for (int t = 0; t < 8; ++t) {
#pragma unroll
            for (int c = 0; c < NUM_CLASSES; ++c) {
                atomicAdd(&sSums[c * DIM + t * 16 + col], acc[t][c]);
            }
        }
    }
    if (col < NUM_CLASSES) atomicAdd(&sCnt[col], cntacc);  // both halves contribute
    __syncthreads();

    for (int i = threadIdx.x; i < NUM_CLASSES * DIM; i += blockDim.x)
        atomicAdd(&ws[i], sSums[i]);
    if (threadIdx.x < NUM_CLASSES)
        atomicAdd(&ws[NUM_CLASSES * DIM + threadIdx.x], sCnt[threadIdx.x]);
}

// Divide by counts, L2-normalize each class center, append target = arange(3).
__global__ __launch_bounds__(DIM) void finalize_kernel(
    const float* __restrict__ ws, float* __restrict__ out)
{
    __shared__ float red[DIM];
    __shared__ float snorm;
    const int d = threadIdx.x;  // 128 threads

    for (int c = 0; c < NUM_CLASSES; ++c) {
        const float cnt    = ws[NUM_CLASSES * DIM + c];
        const float center = ws[c * DIM + d] / cnt;
        red[d] = center * center;
        __syncthreads();
        if (d == 0) {
            float s = 0.0f;
            for (int i = 0; i < DIM; ++i) s += red[i];
            snorm = sqrtf(s);
        }
        __syncthreads();
        out[c * DIM + d] = center / fmaxf(snorm, EPSF);
        __syncthreads();
    }
    if (d < NUM_CLASSES) out[NUM_CLASSES * DIM + d] = (float)d;
}

extern "C" void kernel_launch(void* const* d_in, const int* in_sizes, int n_in,
                              void* d_out, int out_size, void* d_ws, size_t ws_size,
                              hipStream_t stream) {
    const float* features = (const float*)d_in[0];
    const int*   labels   = (const int*)d_in[1];
    float* ws  = (float*)d_ws;
    float* out = (float*)d_out;
    const int N = in_sizes[1];  // number of samples (labels count)

    zero_ws_kernel<<<(WS_FLOATS + 255) / 256, 256, 0, stream>>>(ws);

    // 1024 blocks x 8 waves = 8192 waves; ~15 groups of 4 samples each.
    // Memory-bound: one clean 256 MB streaming pass through features.
    segsum_wmma_kernel<<<1024, 256, 0, stream>>>(features, labels, ws, N);

    finalize_kernel<<<1, DIM, 0, stream>>>(ws, out);
}